// mustang_pooling_36163624632874
// MI455X (gfx1250) — compile-verified
//
#include <hip/hip_runtime.h>
#include <math.h>
#include <float.h>

// ---------------------------------------------------------------------------
// Mustang pooling pipeline for MI455X (gfx1250):
//   3 x [ GCNConv (fp32 WMMA GEMM + edge scatter) -> SAGPool (score GEMV +
//         bitonic top-k) -> stain mean/max pool ]
// GEMM uses V_WMMA_F32_16X16X4_F32 (fp32-exact vs the reference). The whole
// working set (~90MB) is L2-resident (192MB), so the app is memory/latency
// bound and fp32 matrix math costs nothing vs bf16 while matching precision.
// W (64KB) is staged in LDS once per block with an XOR swizzle that makes the
// split-lane B read pattern bank-conflict-free; 8 independent accumulators
// per wave break the WMMA RAW chain.
// ---------------------------------------------------------------------------

#define DFEAT 128
#define NSTAINS 5

typedef __attribute__((ext_vector_type(2))) float v2f;
typedef __attribute__((ext_vector_type(8))) float v8f;

static __device__ __forceinline__ void atomic_max_float(float* addr, float val) {
  // signed-max for non-negative, unsigned-min for negative; init to -FLT_MAX.
  if (val >= 0.0f) atomicMax((int*)addr, __float_as_int(val));
  else             atomicMin((unsigned int*)addr, (unsigned int)__float_as_int(val));
}

// ----------------------------- fill kernels --------------------------------
__global__ void k_fill_f32(float* p, float v, int n) {
  int i = blockIdx.x * blockDim.x + threadIdx.x;
  if (i < n) p[i] = v;
}
__global__ void k_fill_i32(int* p, int v, int n) {
  int i = blockIdx.x * blockDim.x + threadIdx.x;
  if (i < n) p[i] = v;
}

// ------------------------------ edge init ----------------------------------
__global__ void k_init_edges(const int* src_in, const int* dst_in,
                             int* src, int* dst, float* ew, int E) {
  int e = blockIdx.x * blockDim.x + threadIdx.x;
  if (e < E) { src[e] = src_in[e]; dst[e] = dst_in[e]; ew[e] = 1.0f; }
}

// ------------------------------ degree -------------------------------------
__global__ void k_deg(const int* dst, const float* ew, float* deg, int E) {
  int e = blockIdx.x * blockDim.x + threadIdx.x;
  if (e < E) atomicAdd(&deg[dst[e]], ew[e]);
}
__global__ void k_dinv(const float* deg, float* dinv, int n) {
  int i = blockIdx.x * blockDim.x + threadIdx.x;
  if (i < n) {
    float d = deg[i];
    dinv[i] = (d > 0.0f) ? rsqrtf(fmaxf(d, 1e-12f)) : 0.0f;
  }
}

// ------------------- fp32 WMMA GEMM: Y[N,128] = X[N,128] @ W[128,128] -------
// 256-thread block = 8 waves, each wave computes a 16x128 row tile.
// W staged in LDS (64KB, XOR-swizzled for conflict-free split-lane reads).
__global__ void __launch_bounds__(256)
k_gemm_f32_wmma(const float* __restrict__ X, const float* __restrict__ W,
                float* __restrict__ Y, int nRows) {
  __shared__ float ws[DFEAT * DFEAT];          // exactly 64 KB

  // cooperative swizzled load of W: ws[row][col ^ ((row&6)<<3)]
  for (int t = threadIdx.x; t < DFEAT * (DFEAT / 4); t += 256) {
    int row = t >> 5;                          // 32 float4s per row
    int c4  = (t & 31) << 2;
    float4 v = *(const float4*)(W + (size_t)row * DFEAT + c4);
    int swz = (row & 6) << 3;                  // 0/16/32/48
    *(float4*)&ws[row * DFEAT + (c4 ^ swz)] = v;
  }
  __syncthreads();

  int wave = threadIdx.x >> 5;
  int lane = threadIdx.x & 31;
  int half = lane >> 4;                        // 0: lanes 0-15, 1: lanes 16-31
  int idx  = lane & 15;

  int row0 = (blockIdx.x * 8 + wave) * 16;
  int arow = row0 + idx;
  bool rvalid = arow < nRows;
  float rscale = rvalid ? 1.0f : 0.0f;
  const float* xr = X + (size_t)(rvalid ? arow : 0) * DFEAT;

  v8f acc[8] = {};
  for (int k0 = 0; k0 < DFEAT; k0 += 4) {
    int ka = k0 + 2 * half;   // A/B: lanes 0-15 hold K=k0,k0+1; lanes 16-31 K=k0+2,k0+3
    v2f a = *(const v2f*)(xr + ka);            // single b64 load, branch-free
    a[0] *= rscale;
    a[1] *= rscale;
    int swz = (ka & 6) << 3;                   // same for rows ka and ka+1
#pragma unroll
    for (int c = 0; c < 8; ++c) {
      v2f b;
      b[0] = ws[ka * DFEAT + (((c << 4) + idx) ^ swz)];
      b[1] = ws[(ka + 1) * DFEAT + (((c << 4) + idx) ^ swz)];
      acc[c] = __builtin_amdgcn_wmma_f32_16x16x4_f32(
          /*neg_a=*/false, a, /*neg_b=*/false, b,
          /*c_mod=*/(short)0, acc[c], /*reuse_a=*/false, /*reuse_b=*/false);
    }
  }
  // C/D layout: VGPR r -> row r (lanes 0-15) / row 8+r (lanes 16-31)
#pragma unroll
  for (int c = 0; c < 8; ++c)
    for (int r = 0; r < 8; ++r) {
      int row = row0 + r + 8 * half;
      if (row < nRows) Y[(size_t)row * DFEAT + (c << 4) + idx] = acc[c][r];
    }
}

// ---------------- edge scatter: out[dst] += feat[src] * w -------------------
// 4 edges per 128-thread block; 32 lanes x float4 per edge.
__global__ void k_scatter(const int* __restrict__ src, const int* __restrict__ dst,
                          const float* __restrict__ ew, const float* __restrict__ dinv,
                          const float* __restrict__ feat, float* __restrict__ outF,
                          int E, int useNorm) {
  int sub  = threadIdx.x >> 5;
  int lane = threadIdx.x & 31;
  int e = blockIdx.x * 4 + sub;
  if (e >= E) return;
  float w = ew[e];
  if (w == 0.0f) return;
  int s = src[e], t = dst[e];
  if (useNorm) w *= dinv[s] * dinv[t];
  if (w == 0.0f) return;
  float4 v = *(const float4*)(feat + (size_t)s * DFEAT + (lane << 2));
  float* o = outF + (size_t)t * DFEAT + (lane << 2);
  atomicAdd(o + 0, v.x * w);
  atomicAdd(o + 1, v.y * w);
  atomicAdd(o + 2, v.z * w);
  atomicAdd(o + 3, v.w * w);
}

// ------------------------------ bias + relu --------------------------------
__global__ void k_bias_relu(float* h, const float* b, int total) {
  int i = blockIdx.x * blockDim.x + threadIdx.x;
  if (i < total) {
    float v = h[i] + b[i & (DFEAT - 1)];
    h[i] = v > 0.0f ? v : 0.0f;
  }
}

// ---------------- score = tanh(agg@Wn + h@Wr + pb), one wave/node -----------
__global__ void k_score(const float* __restrict__ agg, const float* __restrict__ h,
                        const float* __restrict__ Wn, const float* __restrict__ Wr,
                        const float* __restrict__ pb, float* score, int n) {
  int wavesPerBlock = blockDim.x >> 5;
  int node = blockIdx.x * wavesPerBlock + (threadIdx.x >> 5);
  int lane = threadIdx.x & 31;
  if (node >= n) return;
  float acc = 0.0f;
  for (int d = lane; d < DFEAT; d += 32)
    acc += agg[(size_t)node * DFEAT + d] * Wn[d] + h[(size_t)node * DFEAT + d] * Wr[d];
  for (int off = 16; off > 0; off >>= 1) acc += __shfl_xor(acc, off, 32);
  if (lane == 0) score[node] = tanhf(acc + pb[0]);
}

// ----------------------------- top-k machinery -----------------------------
__global__ void k_make_keys(const float* score, unsigned long long* keys,
                            int n, int npad) {
  int i = blockIdx.x * blockDim.x + threadIdx.x;
  if (i >= npad) return;
  if (i < n) {
    unsigned u = __float_as_uint(score[i]);
    u = (u & 0x80000000u) ? ~u : (u | 0x80000000u);   // order-preserving map
    keys[i] = ((unsigned long long)u << 32) | (unsigned long long)(0xFFFFFFFFu - (unsigned)i);
  } else {
    keys[i] = 0ull;                                    // sinks to the bottom
  }
}
__global__ void k_bitonic(unsigned long long* keys, int kk, int j) {
  int i = blockIdx.x * blockDim.x + threadIdx.x;
  int ixj = i ^ j;
  if (ixj > i) {
    unsigned long long a = keys[i], b = keys[ixj];
    bool desc = ((i & kk) == 0);                       // overall descending
    if (desc ? (a < b) : (a > b)) { keys[i] = b; keys[ixj] = a; }
  }
}
__global__ void k_perm(const unsigned long long* keys, int* perm, int k) {
  int j = blockIdx.x * blockDim.x + threadIdx.x;
  if (j < k) perm[j] = (int)(0xFFFFFFFFu - (unsigned)(keys[j] & 0xFFFFFFFFull));
}
__global__ void k_inv_set(const int* perm, int* inv, int k) {
  int j = blockIdx.x * blockDim.x + threadIdx.x;
  if (j < k) inv[perm[j]] = j;
}

// ------------------- x_new[j] = h[perm[j]] * score[perm[j]] -----------------
__global__ void k_xnew(const float* __restrict__ h, const int* __restrict__ perm,
                       const float* __restrict__ score, float* __restrict__ xnew, int k) {
  int j = blockIdx.x;
  int d = threadIdx.x;
  int p = perm[j];
  xnew[(size_t)j * DFEAT + d] = h[(size_t)p * DFEAT + d] * score[p];
}

// ------------------------------ edge remap ---------------------------------
__global__ void k_edge_remap(const int* srcA, const int* dstA, const float* ewA,
                             const int* inv, int* srcB, int* dstB, float* ewB, int E) {
  int e = blockIdx.x * blockDim.x + threadIdx.x;
  if (e >= E) return;
  int s = srcA[e], t = dstA[e];
  float w = ewA[e];
  int ns = inv[s], nd = inv[t];
  bool valid = (ns >= 0) && (nd >= 0) && (w > 0.0f);
  ewB[e]  = valid ? 1.0f : 0.0f;
  srcB[e] = valid ? ns : 0;
  dstB[e] = valid ? nd : 0;
}

// ------------------------------ stain pooling ------------------------------
__global__ void k_stain_acc(const float* __restrict__ xnew, const int* __restrict__ perm,
                            const float* __restrict__ score, const int* __restrict__ node_attr,
                            float* cnt, float* ssum, float* fsum, float* fmax, int k) {
  int j = blockIdx.x;
  int d = threadIdx.x;
  int p = perm[j];
  int s = node_attr[p];
  if (d == 0) {
    atomicAdd(&cnt[s], 1.0f);
    atomicAdd(&ssum[s], score[p]);
  }
  float v = xnew[(size_t)j * DFEAT + d];
  atomicAdd(&fsum[s * DFEAT + d], v);
  atomic_max_float(&fmax[s * DFEAT + d], v);
}

__global__ void k_stain_final(const float* cnt, const float* ssum,
                              const float* fsum, const float* fmax, float* out) {
  __shared__ float w[NSTAINS];
  __shared__ float c_sh[NSTAINS];
  int d = threadIdx.x;
  if (d == 0) {
    float ms[NSTAINS]; float tot = 0.0f;
    for (int s = 0; s < NSTAINS; ++s) {
      float c = cnt[s];
      c_sh[s] = c;
      ms[s] = (c > 0.0f) ? (ssum[s] / fmaxf(c, 1.0f)) : 0.0f;
      tot += ms[s];
    }
    for (int s = 0; s < NSTAINS; ++s) w[s] = ms[s] / tot;
  }
  __syncthreads();
  float mp = 0.0f, xp = 0.0f;
  for (int s = 0; s < NSTAINS; ++s) {
    float c = c_sh[s];
    float favg = fsum[s * DFEAT + d] / fmaxf(c, 1.0f);
    float fm = (c > 0.0f) ? fmax[s * DFEAT + d] : 0.0f;
    mp += w[s] * favg;
    xp += w[s] * fm;
  }
  out[d] = mp;
  out[DFEAT + d] = xp;
}

// ---------------------------------------------------------------------------
static inline int npow2(int n) { int p = 1; while (p < n) p <<= 1; return p; }
static inline int cdiv(int a, int b) { return (a + b - 1) / b; }

extern "C" void kernel_launch(void* const* d_in, const int* in_sizes, int n_in,
                              void* d_out, int out_size, void* d_ws, size_t ws_size,
                              hipStream_t stream) {
  (void)in_sizes; (void)n_in; (void)out_size; (void)ws_size;
  const int N = 50000, E = 600000, L = 3;

  const float* x_in    = (const float*)d_in[0];
  const float* conv_W  = (const float*)d_in[1];
  const float* conv_b  = (const float*)d_in[2];
  const float* pool_Wn = (const float*)d_in[3];
  const float* pool_Wr = (const float*)d_in[4];
  const float* pool_b  = (const float*)d_in[5];
  const int*   src_in  = (const int*)d_in[6];
  const int*   dst_in  = (const int*)d_in[7];
  const int*   nattr   = (const int*)d_in[8];
  float* out = (float*)d_out;

  // ---- workspace carve-up (aligned to 256B) ----
  char* base = (char*)d_ws;
  size_t off = 0;
  auto carve = [&](size_t bytes) -> char* {
    char* p = base + off;
    off += (bytes + 255) & ~(size_t)255;
    return p;
  };
  float* bufX = (float*)carve((size_t)N * DFEAT * 4);  // current node features
  float* bufT = (float*)carve((size_t)N * DFEAT * 4);  // xw, then agg
  float* bufH = (float*)carve((size_t)N * DFEAT * 4);  // conv output h
  float* deg  = (float*)carve((size_t)N * 4);
  float* dinv = (float*)carve((size_t)N * 4);
  float* scr  = (float*)carve((size_t)N * 4);
  unsigned long long* keys = (unsigned long long*)carve((size_t)65536 * 8);
  int* perm = (int*)carve((size_t)N * 4);
  int* inv  = (int*)carve((size_t)N * 4);
  int*   srcA = (int*)carve((size_t)E * 4);
  int*   dstA = (int*)carve((size_t)E * 4);
  float* ewA  = (float*)carve((size_t)E * 4);
  int*   srcB = (int*)carve((size_t)E * 4);
  int*   dstB = (int*)carve((size_t)E * 4);
  float* ewB  = (float*)carve((size_t)E * 4);
  float* cnt  = (float*)carve(NSTAINS * 4);
  float* ssum = (float*)carve(NSTAINS * 4);
  float* fsum = (float*)carve(NSTAINS * DFEAT * 4);
  float* fmax = (float*)carve(NSTAINS * DFEAT * 4);

  k_init_edges<<<cdiv(E, 256), 256, 0, stream>>>(src_in, dst_in, srcA, dstA, ewA, E);

  const float* xcur = x_in;
  int* cs = srcA; int* cd = dstA; float* cw = ewA;
  int* ns = srcB; int* nd = dstB; float* nw = ewB;
  int Ncur = N;

  for (int i = 0; i < L; ++i) {
    int k = (Ncur + 1) / 2;                       // ceil(0.5 * N)

    // degree + symmetric normalization
    k_fill_f32<<<cdiv(Ncur, 256), 256, 0, stream>>>(deg, 0.0f, Ncur);
    k_deg<<<cdiv(E, 256), 256, 0, stream>>>(cd, cw, deg, E);
    k_dinv<<<cdiv(Ncur, 256), 256, 0, stream>>>(deg, dinv, Ncur);

    // xw = x @ W_i  (fp32 WMMA, 8 waves/block, 16x128 tile per wave)
    k_gemm_f32_wmma<<<cdiv(Ncur, 128), 256, 0, stream>>>(
        xcur, conv_W + (size_t)i * DFEAT * DFEAT, bufT, Ncur);

    // h = relu(scatter(xw[src]*norm -> dst) + b)
    k_fill_f32<<<cdiv(Ncur * DFEAT, 256), 256, 0, stream>>>(bufH, 0.0f, Ncur * DFEAT);
    k_scatter<<<cdiv(E, 4), 128, 0, stream>>>(cs, cd, cw, dinv, bufT, bufH, E, 1);
    k_bias_relu<<<cdiv(Ncur * DFEAT, 256), 256, 0, stream>>>(
        bufH, conv_b + (size_t)i * DFEAT, Ncur * DFEAT);

    // agg = scatter(h[src]*ew -> dst); score = tanh(agg@Wn + h@Wr + pb)
    k_fill_f32<<<cdiv(Ncur * DFEAT, 256), 256, 0, stream>>>(bufT, 0.0f, Ncur * DFEAT);
    k_scatter<<<cdiv(E, 4), 128, 0, stream>>>(cs, cd, cw, dinv, bufH, bufT, E, 0);
    k_score<<<cdiv(Ncur, 8), 256, 0, stream>>>(
        bufT, bufH, pool_Wn + (size_t)i * DFEAT, pool_Wr + (size_t)i * DFEAT,
        pool_b + i, scr, Ncur);

    // top-k via descending bitonic sort of packed keys
    int npad = npow2(Ncur);
    k_make_keys<<<cdiv(npad, 256), 256, 0, stream>>>(scr, keys, Ncur, npad);
    for (int kk = 2; kk <= npad; kk <<= 1)
      for (int j = kk >> 1; j > 0; j >>= 1)
        k_bitonic<<<npad / 256, 256, 0, stream>>>(keys, kk, j);
    k_perm<<<cdiv(k, 256), 256, 0, stream>>>(keys, perm, k);

    // inverse map + filter_adj + gated features
    k_fill_i32<<<cdiv(Ncur, 256), 256, 0, stream>>>(inv, -1, Ncur);
    k_inv_set<<<cdiv(k, 256), 256, 0, stream>>>(perm, inv, k);
    k_xnew<<<k, DFEAT, 0, stream>>>(bufH, perm, scr, bufX, k);
    k_edge_remap<<<cdiv(E, 256), 256, 0, stream>>>(cs, cd, cw, inv, ns, nd, nw, E);

    // stain pooling -> out[i*256 .. i*256+255]
    k_fill_f32<<<1, 256, 0, stream>>>(cnt, 0.0f, NSTAINS);
    k_fill_f32<<<1, 256, 0, stream>>>(ssum, 0.0f, NSTAINS);
    k_fill_f32<<<cdiv(NSTAINS * DFEAT, 256), 256, 0, stream>>>(fsum, 0.0f, NSTAINS * DFEAT);
    k_fill_f32<<<cdiv(NSTAINS * DFEAT, 256), 256, 0, stream>>>(fmax, -FLT_MAX, NSTAINS * DFEAT);
    k_stain_acc<<<k, DFEAT, 0, stream>>>(bufX, perm, scr, nattr, cnt, ssum, fsum, fmax, k);
    k_stain_final<<<1, DFEAT, 0, stream>>>(cnt, ssum, fsum, fmax, out + (size_t)i * 2 * DFEAT);

    // rotate for next layer
    xcur = bufX;
    int* ts;  float* tf;
    ts = cs; cs = ns; ns = ts;
    ts = cd; cd = nd; nd = ts;
    tf = cw; cw = nw; nw = tf;
    Ncur = k;
  }
}